// LSTM_17042430230612
// MI455X (gfx1250) — compile-verified
//
#include <hip/hip_runtime.h>

// ---------------------------------------------------------------------------
// LSTM (B=4096,T=512,I=15,H=16) -> only the LAST batch element's hidden
// sequence feeds the output, so we drop 4095/4096 of the reference work.
//
// Phase 1: pre[t][g] = x[last,t,:]·W_ih[g,:] + b_ih[g] + b_hh[g]
//          [512x15]x[15x64] f32 GEMM via V_WMMA_F32_16X16X4_F32 (exact f32),
//          one wave per 16x16 tile, 4 chained K=4 WMMAs (K padded 15->16).
//          Padding handled branch-free: only K=15 (kc==3, upper half-wave)
//          is OOB -> clamp address, v_cndmask the value to zero.
// Phase 2: single-wave sequential scan over T=512 (latency-bound chain),
//          W_hh rows in registers, h in LDS, out[t]=W_reg·h+b_reg via shfl.
// ---------------------------------------------------------------------------

typedef float v2f __attribute__((ext_vector_type(2)));
typedef float v8f __attribute__((ext_vector_type(8)));

#define T_LEN 512
#define I_DIM 15
#define G_DIM 64   // 4*H

// ---- Phase 1: WMMA GEMM for the input contribution -------------------------
__global__ __launch_bounds__(32) void lstm_xg_wmma(
    const float* __restrict__ x,      // [4096,512,15]
    const float* __restrict__ Wih,    // [64,15] row-major
    const float* __restrict__ bih,    // [64]
    const float* __restrict__ bhh,    // [64]
    float* __restrict__ pre)          // [512,64]
{
    const int lane  = threadIdx.x;            // wave32, EXEC all ones
    const int t0    = blockIdx.x * 16;        // timestep tile
    const int g0    = blockIdx.y * 16;        // gate tile
    const int mn    = lane & 15;              // M (for A/D) == N (for B/D)
    const int kLane = (lane >> 4) << 1;       // lanes 0-15 -> K{0,1}; 16-31 -> K{2,3}

    const float* xb = x + (size_t)4095 * T_LEN * I_DIM;  // last batch element
    const int aRow = (t0 + mn) * I_DIM;       // row base into x (A fragment)
    const int bRow = (g0 + mn) * I_DIM;       // row base into W_ih (B fragment)

    v8f acc = {};

    // Chunks 0..2: K in [0,11], all indices in-bounds -> straight-line loads.
#pragma unroll
    for (int kc = 0; kc < 3; ++kc) {
        const int k0 = kc * 4 + kLane;
        v2f a, b;
        a.x = xb [aRow + k0];
        a.y = xb [aRow + k0 + 1];
        b.x = Wih[bRow + k0];
        b.y = Wih[bRow + k0 + 1];
        acc = __builtin_amdgcn_wmma_f32_16x16x4_f32(
            false, a, false, b, (short)0, acc, false, false);
    }

    // Chunk 3: K in {12..15}; only k==15 (upper half-wave's .y) is padding.
    {
        const int  k0   = 12 + kLane;         // 12 or 14, always in-bounds
        const bool pad  = (kLane == 2);       // k0+1 == 15 -> zero element
        const int  k1c  = pad ? 14 : (k0 + 1);// clamped: load stays in-bounds
        v2f a, b;
        a.x = xb [aRow + k0];
        b.x = Wih[bRow + k0];
        const float ay = xb [aRow + k1c];
        const float by = Wih[bRow + k1c];
        a.y = pad ? 0.0f : ay;                // v_cndmask, no branch
        b.y = pad ? 0.0f : by;
        acc = __builtin_amdgcn_wmma_f32_16x16x4_f32(
            false, a, false, b, (short)0, acc, false, false);
    }

    const int   col   = g0 + mn;
    const float bias  = bih[col] + bhh[col];
    const int   rbase = (lane < 16) ? 0 : 8;  // D: VGPR v -> row v (+8 upper half)
#pragma unroll
    for (int v = 0; v < 8; ++v) {
        const int t = t0 + rbase + v;
        pre[t * G_DIM + col] = acc[v] + bias;
    }
}

// ---- Phase 2: single-wave sequential LSTM scan -----------------------------
__global__ __launch_bounds__(32) void lstm_scan(
    const float* __restrict__ pre,    // [512,64]
    const float* __restrict__ Whh,    // [64,16] row-major
    const float* __restrict__ Wreg,   // [16]
    const float* __restrict__ breg,   // [1]
    float* __restrict__ out)          // [512]
{
    const int l = threadIdx.x;        // 0..31
    const int j = l & 15;

    // Each lane owns gate rows l and l+32 of W_hh in registers.
    float w0[16], w1[16];
#pragma unroll
    for (int k = 0; k < 16; ++k) {
        w0[k] = Whh[l * 16 + k];
        w1[k] = Whh[(l + 32) * 16 + k];
    }
    const float wr = Wreg[j];
    const float br = breg[0];

    __shared__ float hs[16];
    __shared__ float gI[16], gF[16], gG[16], gO[16];
    if (l < 16) hs[l] = 0.0f;
    __syncthreads();

    float c = 0.0f;   // lanes 16-31 shadow-compute the identical recurrence
    for (int t = 0; t < T_LEN; ++t) {
        float acc0 = pre[t * G_DIM + l];
        float acc1 = pre[t * G_DIM + l + 32];
#pragma unroll
        for (int k = 0; k < 16; ++k) {
            acc0 = fmaf(w0[k], hs[k], acc0);
            acc1 = fmaf(w1[k], hs[k], acc1);
        }
        // PyTorch gate order i,f,g,o:
        //  lanes 0-15 : acc0 = i-gate, acc1 = g-gate
        //  lanes 16-31: acc0 = f-gate, acc1 = o-gate
        if (l < 16) {
            gI[j] = 1.0f / (1.0f + __expf(-acc0));
            gG[j] = tanhf(acc1);
        } else {
            gF[j] = 1.0f / (1.0f + __expf(-acc0));
            gO[j] = 1.0f / (1.0f + __expf(-acc1));
        }
        __syncthreads();

        c = fmaf(gF[j], c, gI[j] * gG[j]);
        const float h = gO[j] * tanhf(c);
        if (l < 16) hs[j] = h;

        // out[t] = W_reg . h + b_reg  (lanes 16-31 contribute 0)
        float partial = (l < 16) ? wr * h : 0.0f;
#pragma unroll
        for (int off = 16; off >= 1; off >>= 1)
            partial += __shfl_xor(partial, off, 32);
        if (l == 0) out[t] = partial + br;
        __syncthreads();  // hs visible before next step's matvec
    }
}

// ---------------------------------------------------------------------------
extern "C" void kernel_launch(void* const* d_in, const int* in_sizes, int n_in,
                              void* d_out, int out_size, void* d_ws, size_t ws_size,
                              hipStream_t stream) {
    (void)in_sizes; (void)n_in; (void)out_size; (void)ws_size;
    const float* x    = (const float*)d_in[0];
    const float* Wih  = (const float*)d_in[1];
    const float* Whh  = (const float*)d_in[2];
    const float* bih  = (const float*)d_in[3];
    const float* bhh  = (const float*)d_in[4];
    const float* Wreg = (const float*)d_in[5];
    const float* breg = (const float*)d_in[6];
    float* out = (float*)d_out;
    float* pre = (float*)d_ws;   // [512][64] f32 = 128 KB scratch (fits in L2)

    dim3 gridA(T_LEN / 16, G_DIM / 16);   // 32 x 4 tiles
    lstm_xg_wmma<<<gridA, 32, 0, stream>>>(x, Wih, bih, bhh, pre);
    lstm_scan<<<1, 32, 0, stream>>>(pre, Whh, Wreg, breg, out);
}